// CostVolumeRight_53901839565186
// MI455X (gfx1250) — compile-verified
//
#include <hip/hip_runtime.h>

// Problem constants (match reference)
#define B_ 8
#define C_ 256
#define H_ 160
#define W_ 320
#define D_ 64

#define KC 32               // channels per LDS chunk
#define NCHUNK (C_ / KC)    // 8
#define LSTRIDE 336         // LDS row stride in floats; 336 % 32 == 16 -> conflict-free frags
#define NTHREADS 640        // 20 waves (wave32); one 16-wide x-tile per wave -> 20*16 = W

typedef __attribute__((ext_vector_type(2))) float v2f;
typedef __attribute__((ext_vector_type(4))) float v4f;
typedef __attribute__((ext_vector_type(8))) float v8f;
typedef __attribute__((ext_vector_type(4))) int   v4i;

// ---- optional CDNA5 async global->LDS path (ASYNCcnt), guarded so the file
// ---- always compiles; fallback is reg-staging + ds_store_b128.
#if defined(__has_builtin)
#if __has_builtin(__builtin_amdgcn_global_load_async_to_lds_b128)
#define USE_ASYNC_LDS 1
#endif
#endif
#ifndef USE_ASYNC_LDS
#define USE_ASYNC_LDS 0
#endif

__device__ __forceinline__ void wait_async0() {
#if defined(__has_builtin) && __has_builtin(__builtin_amdgcn_s_wait_asynccnt)
    __builtin_amdgcn_s_wait_asynccnt(0);
#else
    asm volatile("s_wait_asynccnt 0x0" ::: "memory");
#endif
}

__global__ __launch_bounds__(NTHREADS, 1)
void corr_wmma_f32_kernel(const float* __restrict__ left,
                          const float* __restrict__ right,
                          float* __restrict__ out)
{
    // 168 KB static LDS: double-buffered L and R chunks [KC][LSTRIDE]
    __shared__ float s_lds[4 * KC * LSTRIDE];

    // buffer selection by integer offset only -- no static arrays of
    // LDS-derived pointers (lld can't relocate addrspacecast initializers)
    auto Lptr = [&](int p) -> float* { return s_lds + p * (KC * LSTRIDE); };
    auto Rptr = [&](int p) -> float* { return s_lds + 2 * (KC * LSTRIDE) + p * (KC * LSTRIDE); };

    const int by   = blockIdx.x;           // b*H + y
    const int b    = by / H_;
    const int y    = by - b * H_;
    const int tid  = threadIdx.x;
    const int wave = tid >> 5;             // wave32
    const int lane = tid & 31;
    const int hf   = lane >> 4;            // half-wave: selects K pair / M+8
    const int ln   = lane & 15;
    const int X0   = wave << 4;            // this wave's x-tile origin

    const size_t CHAN = (size_t)H_ * W_;
    const size_t base = (size_t)b * C_ * CHAN + (size_t)y * W_;

    // valid x'-tile count for the (cold) scatter store; hot loop is unconditional
    int jlim = (W_ - X0) >> 4;
    if (jlim > 5) jlim = 5;

    v8f acc[5];
#pragma unroll
    for (int j = 0; j < 5; ++j) { v8f z = {}; acc[j] = z; }

#if USE_ASYNC_LDS
    // ---- CDNA5 async copy: global -> LDS directly, tracked by ASYNCcnt ----
    auto async_load_chunk = [&](int c0, float* Lp, float* Rp) {
#pragma unroll
        for (int r = 0; r < 4; ++r) {
            int i  = tid + NTHREADS * r;       // 0..2559 float4 slots
            int cc = i / 80;                   // channel within chunk (W/4 == 80)
            int g  = i - cc * 80;
            size_t goff = base + (size_t)(c0 + cc) * CHAN + (size_t)(g << 2);
            int    loff = cc * LSTRIDE + (g << 2);
            __builtin_amdgcn_global_load_async_to_lds_b128(
                (v4i*)(left + goff), (v4i*)(Lp + loff), 0, 0);
            __builtin_amdgcn_global_load_async_to_lds_b128(
                (v4i*)(right + goff), (v4i*)(Rp + loff), 0, 0);
        }
    };
#else
    // ---- fallback: global -> VGPR -> ds_store_b128 ----
    v4f lreg[4], rreg[4];
    auto load_chunk = [&](int c0) {
#pragma unroll
        for (int r = 0; r < 4; ++r) {
            int i  = tid + NTHREADS * r;
            int cc = i / 80;
            int g  = i - cc * 80;
            size_t off = base + (size_t)(c0 + cc) * CHAN + (size_t)(g << 2);
            lreg[r] = *(const v4f*)(left  + off);
            rreg[r] = *(const v4f*)(right + off);
        }
    };
    auto store_chunk = [&](float* Lp, float* Rp) {
#pragma unroll
        for (int r = 0; r < 4; ++r) {
            int i  = tid + NTHREADS * r;
            int cc = i / 80;
            int g  = i - cc * 80;
            int off = cc * LSTRIDE + (g << 2);
            *(v4f*)(Lp + off) = lreg[r];
            *(v4f*)(Rp + off) = rreg[r];
        }
    };
#endif

    // ---- branch-free WMMA compute over one staged K-chunk ----
    // All offsets are compile-time constants relative to two per-chunk bases,
    // so every ds_load uses an immediate offset and EXEC is never touched.
    auto compute_chunk = [&](const float* Lp, const float* Rp) {
        const float* Ab = Lp + 2 * hf * LSTRIDE + X0 + ln;  // A: L[c, X0+16j+m]
        const float* Bb = Rp + 2 * hf * LSTRIDE + X0 + ln;  // B: R[c, X0+n]
#pragma unroll
        for (int ks = 0; ks < KC / 4; ++ks) {
            const int rowo = 4 * ks * LSTRIDE;
            v2f bfrag;
            bfrag.x = Bb[rowo];
            bfrag.y = Bb[rowo + LSTRIDE];
#pragma unroll
            for (int j = 0; j < 5; ++j) {
                v2f afrag;
                afrag.x = Ab[rowo + (j << 4)];
                afrag.y = Ab[rowo + LSTRIDE + (j << 4)];
                acc[j] = __builtin_amdgcn_wmma_f32_16x16x4_f32(
                    false, afrag, false, bfrag, (short)0, acc[j], false, false);
            }
        }
    };

    // ---- software-pipelined main loop ----
#if USE_ASYNC_LDS
    async_load_chunk(0, Lptr(0), Rptr(0));
    wait_async0();
    __syncthreads();
    for (int ch = 0; ch < NCHUNK; ++ch) {
        const bool last = (ch == NCHUNK - 1);
        const int p = ch & 1;
        if (!last) async_load_chunk((ch + 1) * KC, Lptr(1 - p), Rptr(1 - p));
        compute_chunk(Lptr(p), Rptr(p));
        if (!last) {
            wait_async0();       // own async copies have landed in LDS
            __syncthreads();     // everyone's copies landed; everyone done with buf[p]
        }
    }
#else
    load_chunk(0);
    store_chunk(Lptr(0), Rptr(0));
    __syncthreads();
    for (int ch = 0; ch < NCHUNK; ++ch) {
        const bool last = (ch == NCHUNK - 1);
        const int p = ch & 1;
        if (!last) load_chunk((ch + 1) * KC);   // global loads in flight over compute
        compute_chunk(Lptr(p), Rptr(p));
        __syncthreads();
        if (!last) {
            store_chunk(Lptr(1 - p), Rptr(1 - p));
            __syncthreads();
        }
    }
#endif

    // ---- band extraction + scatter store: d = 16j + M - N, x = X0 + N ----
    const float scale = 1.0f / (float)C_;       // mean over channels
    const int xo = X0 + ln;
#pragma unroll
    for (int j = 0; j < 5; ++j) {
        if (j < jlim) {
#pragma unroll
            for (int v = 0; v < 8; ++v) {
                const int M = v + (hf << 3);    // C/D layout: VGPR v -> M=v | v+8
                const int d = (j << 4) + M - ln;
                if (d >= 0 && d < D_) {
                    out[(((size_t)b * D_ + d) * H_ + y) * W_ + xo] = acc[j][v] * scale;
                }
            }
        }
    }
}

extern "C" void kernel_launch(void* const* d_in, const int* in_sizes, int n_in,
                              void* d_out, int out_size, void* d_ws, size_t ws_size,
                              hipStream_t stream) {
    const float* left  = (const float*)d_in[0];
    const float* right = (const float*)d_in[1];
    float* out = (float*)d_out;

    // boundary-clipped columns must be zero; scatter only writes the valid band
    (void)hipMemsetAsync(out, 0, (size_t)out_size * sizeof(float), stream);

    dim3 grid(B_ * H_);      // one workgroup per (b, y) row
    dim3 block(NTHREADS);    // 20 waves: one 16-wide x-tile each
    corr_wmma_f32_kernel<<<grid, block, 0, stream>>>(left, right, out);
}